// Attention_80367428043370
// MI455X (gfx1250) — compile-verified
//
#include <hip/hip_runtime.h>

typedef __attribute__((ext_vector_type(2))) float v2f;
typedef __attribute__((ext_vector_type(8))) float v8f;

#define Hh 64
#define Ww 64
#define Cc 256
#define Rt 16              // output rows per block
#define CHB 32             // output channels per block (2 WMMA sub-tiles)
#define LROWS 18           // Rt + 2 halo rows
#define LCOLS 66           // Ww + 2 halo cols
#define BLOCK 512          // 16 waves -> 4 waves per SIMD32 (1 WG/WGP due to LDS)
#define NWAVES (BLOCK / 32)
#define WLF (CHB * Cc)            // W tile floats   (8192  -> 32 KB)
#define QTF (CHB * LROWS * LCOLS) // q tile floats   (38016 -> 152 KB)
#define RELF (CHB * 9)            // rel tile floats (288)

__global__ __launch_bounds__(BLOCK) void fused_neigh_attn(
    const float* __restrict__ x,    // (B, 256, 64, 64)
    const float* __restrict__ Wm,   // (256, 256)
    const float* __restrict__ rel,  // (256, 3, 3)
    float* __restrict__ out)        // (B, 256, 64, 64)
{
    extern __shared__ float smem[];
    float* Wl   = smem;               // [CHB][256]
    float* qt   = smem + WLF;         // [CHB][LROWS][LCOLS], cols 0/65 = zero halo
    float* relL = smem + WLF + QTF;   // [CHB][9]

    const int tid = threadIdx.x;
    const int b   = blockIdx.z;
    const int o0  = blockIdx.y * CHB;
    const int h0  = blockIdx.x * Rt;

    // ---- stage W tile, rel tile into LDS; zero the w-halo columns ----
    for (int i = tid; i < WLF; i += BLOCK)
        Wl[i] = Wm[(o0 + (i >> 8)) * Cc + (i & 255)];
    if (tid < RELF)
        relL[tid] = rel[o0 * 9 + tid];
    for (int i = tid; i < CHB * LROWS; i += BLOCK) {
        qt[i * LCOLS + 0]  = 0.f;
        qt[i * LCOLS + 65] = 0.f;
    }
    __syncthreads();

    // ---- phase 1: q = W * x GEMM into LDS via V_WMMA_F32_16X16X4_F32 ----
    // Force the wave index scalar so tile/halo branches are s_cbranch (EXEC
    // stays all-ones through the WMMA region, as the ISA requires).
    const int wave = __builtin_amdgcn_readfirstlane(tid >> 5);
    const int lane = tid & 31;
    const int m    = lane & 15;          // A row / B-C-D column within tile
    const int koff = (lane >> 4) << 1;   // lanes 0-15: K{0,1}; lanes 16-31: K{2,3}
    const int chh  = (lane >> 4) << 3;   // C/D: VGPR r holds rows r (lo) / r+8 (hi)

    // 144 N-tiles: 2 channel sub-tiles x 18 halo rows x 4 column groups of 16
    for (int t = wave; t < 2 * LROWS * 4; t += NWAVES) {
        const int ot = t / (LROWS * 4);
        const int rr = (t - ot * (LROWS * 4)) >> 2;
        const int g  = t & 3;
        const int h  = h0 - 1 + rr;          // source row in the image
        const int w  = (g << 4) + m;
        float* qcol = &qt[(ot * 16) * (LROWS * LCOLS) + rr * LCOLS + 1 + w];

        if (h < 0 || h >= Hh) {              // zero halo row (wave-uniform)
#pragma unroll
            for (int r = 0; r < 8; ++r)
                qcol[(chh + r) * (LROWS * LCOLS)] = 0.f;
            continue;
        }

        const float* xb = x + ((size_t)b * Cc + koff) * (Hh * Ww) + h * Ww + w;
        const float* wl = &Wl[((ot << 4) + m) * Cc + koff];

        v8f c = {};
#pragma unroll 8
        for (int k = 0; k < Cc / 4; ++k) {
            v2f a = *(const v2f*)(wl + 4 * k);            // ds_load_b64
            v2f bv;
            bv.x = xb[(size_t)(4 * k + 0) * (Hh * Ww)];   // coalesced dword loads
            bv.y = xb[(size_t)(4 * k + 1) * (Hh * Ww)];
            c = __builtin_amdgcn_wmma_f32_16x16x4_f32(
                    false, a, false, bv, (short)0, c, false, false);
        }
#pragma unroll
        for (int r = 0; r < 8; ++r)
            qcol[(chh + r) * (LROWS * LCOLS)] = c[r];
    }
    __syncthreads();

    // ---- phase 2: per-element row-softmax over 3x3 neighborhood (diagonal) ----
    const size_t obase = ((size_t)b * Cc + o0) * (Hh * Ww);
    for (int e = tid; e < CHB * Rt * Ww; e += BLOCK) {
        const int ch  = e >> 10;         // / (Rt*Ww)
        const int rem = e & 1023;
        const int r   = rem >> 6;
        const int w   = rem & 63;
        const float* qch = &qt[ch * (LROWS * LCOLS)];
        const float  qc  = qch[(r + 1) * LCOLS + (w + 1)];
        const float* rl  = &relL[ch * 9];

        float acc = 0.f;
#pragma unroll
        for (int ki = 0; ki < 3; ++ki) {
            const float* row = &qch[(r + ki) * LCOLS + w];
            const float v0 = row[0], v1 = row[1], v2 = row[2];
            const float l0 = qc * (v0 + rl[ki * 3 + 0]);
            const float l1 = qc * (v1 + rl[ki * 3 + 1]);
            const float l2 = qc * (v2 + rl[ki * 3 + 2]);
            const float mx = fmaxf(l0, fmaxf(l1, l2));
            const float e0 = __expf(l0 - mx);
            const float e1 = __expf(l1 - mx);
            const float e2 = __expf(l2 - mx);
            const float den = e0 + e1 + e2;
            const float vd = (ki == 0) ? v0 : (ki == 1) ? v1 : v2;
            const float ed = (ki == 0) ? e0 : (ki == 1) ? e1 : e2;
            acc += vd * ed / den;
        }
        out[obase + (size_t)ch * (Hh * Ww) + (size_t)(h0 + r) * Ww + w] = acc;
    }
}

extern "C" void kernel_launch(void* const* d_in, const int* in_sizes, int n_in,
                              void* d_out, int out_size, void* d_ws, size_t ws_size,
                              hipStream_t stream) {
    const float* x   = (const float*)d_in[0];
    const float* Wm  = (const float*)d_in[1];
    const float* rel = (const float*)d_in[2];
    float* out = (float*)d_out;

    const int B = in_sizes[0] / (Cc * Hh * Ww);   // 8
    dim3 grid(Hh / Rt, Cc / CHB, B);              // (4, 8, 8) = 256 blocks
    const size_t shmem = (size_t)(WLF + QTF + RELF) * sizeof(float); // ~186 KB < 320 KB WGP LDS
    fused_neigh_attn<<<grid, dim3(BLOCK), shmem, stream>>>(x, Wm, rel, out);
}